// StructuralTypeSeqModel_52553219834608
// MI455X (gfx1250) — compile-verified
//
#include <hip/hip_runtime.h>
#include <stdint.h>

#define N_NODES 49920
#define E_EDGES 600000
#define B_GRAPHS 64
#define F_INDIM 32
#define H_DIM 128
#define HOPS 5
#define NUM_CL 24
#define SEQ_L 20
#define NEG_SLOPE 0.2f
#define NODES_PER_GRAPH (N_NODES / B_GRAPHS)   /* 780 */
#define OUT_PER_GRAPH (NUM_CL * SEQ_L)         /* 480 */

typedef __attribute__((ext_vector_type(2))) float v2f;
typedef __attribute__((ext_vector_type(8))) float v8f;

__device__ __forceinline__ float lrelu(float x) {
    return x > 0.0f ? x : NEG_SLOPE * x;
}

// ---------------------------------------------------------------------------
// CSR construction (destination-sorted adjacency), deterministic after sort
// ---------------------------------------------------------------------------

__global__ void k_count(const long long* __restrict__ ei, int* __restrict__ cnt) {
    int e = blockIdx.x * blockDim.x + threadIdx.x;
    if (e < E_EDGES) {
        int d = (int)ei[E_EDGES + e];
        atomicAdd(&cnt[d], 1);
    }
}

__global__ void k_scan_block(const int* __restrict__ in, int* __restrict__ out,
                             int* __restrict__ bsum) {
    __shared__ int s[256];
    int i = blockIdx.x * 256 + threadIdx.x;
    int v = in[i];
    s[threadIdx.x] = v;
    __syncthreads();
    for (int d = 1; d < 256; d <<= 1) {
        int t = (threadIdx.x >= d) ? s[threadIdx.x - d] : 0;
        __syncthreads();
        s[threadIdx.x] += t;
        __syncthreads();
    }
    out[i] = s[threadIdx.x] - v;               // exclusive
    if (threadIdx.x == 255) bsum[blockIdx.x] = s[255];
}

__global__ void k_scan_bsum(int* __restrict__ bsum, int nb) {
    __shared__ int s[256];
    int v = (threadIdx.x < nb) ? bsum[threadIdx.x] : 0;
    s[threadIdx.x] = v;
    __syncthreads();
    for (int d = 1; d < 256; d <<= 1) {
        int t = (threadIdx.x >= d) ? s[threadIdx.x - d] : 0;
        __syncthreads();
        s[threadIdx.x] += t;
        __syncthreads();
    }
    if (threadIdx.x < nb) bsum[threadIdx.x] = s[threadIdx.x] - v;  // exclusive
}

__global__ void k_scan_add(int* __restrict__ out, const int* __restrict__ bsum) {
    int i = blockIdx.x * 256 + threadIdx.x;
    out[i] += bsum[blockIdx.x];
}

__global__ void k_fill(const long long* __restrict__ ei, const int* __restrict__ offs,
                       int* __restrict__ cursor, int* __restrict__ csr) {
    int e = blockIdx.x * blockDim.x + threadIdx.x;
    if (e < E_EDGES) {
        int s = (int)ei[e];
        int d = (int)ei[E_EDGES + e];
        int pos = offs[d] + atomicAdd(&cursor[d], 1);
        csr[pos] = s;
    }
}

// per-node insertion sort -> deterministic adjacency order (duplicates are
// identical contributions, so any tie order gives identical bits)
__global__ void k_sortlists(const int* __restrict__ offs, const int* __restrict__ cnt,
                            int* __restrict__ csr) {
    int n = blockIdx.x * blockDim.x + threadIdx.x;
    if (n >= N_NODES) return;
    int st = offs[n], deg = cnt[n];
    for (int i = 1; i < deg; ++i) {
        int key = csr[st + i];
        int j = i - 1;
        while (j >= 0 && csr[st + j] > key) { csr[st + j + 1] = csr[st + j]; --j; }
        csr[st + j + 1] = key;
    }
}

// ---------------------------------------------------------------------------
// Dense GEMM  Hout[N,128] = X[N,K] @ W[K,128]  via V_WMMA_F32_16X16X4_F32.
// The 16xK A tile is staged once per block into LDS with the CDNA5 async
// copy path (GLOBAL_LOAD_ASYNC_TO_LDS_B128 / ASYNCcnt / s_wait_asynccnt),
// then shared by all 8 column-tile waves (8x reuse, ds_load_b64 fragments).
// VGPR layouts per CDNA5 ISA 7.12.2 (fp32 A 16x4, B 4x16, C/D 16x16).
// ---------------------------------------------------------------------------
__global__ void k_gemm_wmma(const float* __restrict__ X, const float* __restrict__ W,
                            float* __restrict__ Hout, int K) {
    __shared__ float sA[16 * H_DIM];       // up to 8 KB (K <= 128)
    int tid  = threadIdx.x;
    int lane = tid & 31;
    int wave = tid >> 5;                   // 0..7 -> column tile
    int row0 = blockIdx.x * 16;
    int col0 = wave * 16;

    // ---- async stage of A tile: 16*K floats, 16B per lane per iteration ----
    // generic pointer to __shared__ carries the LDS byte offset in addr[31:0]
    uint32_t lds_base = (uint32_t)(uintptr_t)(void*)sA;
    int nvec4 = (16 * K) >> 2;             // 512 (K=128) or 128 (K=32)
    for (int i = tid; i < nvec4; i += 256) {
        uint64_t gaddr = (uint64_t)(uintptr_t)(X + (size_t)row0 * K + (size_t)i * 4);
        uint32_t laddr = lds_base + (uint32_t)i * 16u;
        asm volatile("global_load_async_to_lds_b128 %0, %1, off"
                     :: "v"(laddr), "v"(gaddr) : "memory");
    }
    asm volatile("s_wait_asynccnt 0" ::: "memory");
    __syncthreads();

    int hi = lane >> 4;                    // 0: lanes 0-15, 1: lanes 16-31
    int lo = lane & 15;

    // A fragment source: lane holds row lo; VGPR pair = K {k0,k0+1} / {k0+2,k0+3}
    const float* arow = sA + lo * K + 2 * hi;
    // B: lane holds column (col0+lo); VGPR pair = rows K {0,1} or {2,3} of step
    const float* bcol = W + (size_t)(2 * hi) * H_DIM + col0 + lo;

    v8f c = {};
    for (int k0 = 0; k0 < K; k0 += 4) {
        v2f a, b;
        a.x = arow[k0];
        a.y = arow[k0 + 1];
        b.x = bcol[(size_t)k0 * H_DIM];
        b.y = bcol[(size_t)(k0 + 1) * H_DIM];
        c = __builtin_amdgcn_wmma_f32_16x16x4_f32(
                /*neg_a=*/false, a, /*neg_b=*/false, b,
                /*c_mod=*/(short)0, c, /*reuse_a=*/false, /*reuse_b=*/false);
    }

    // D: VGPR r holds row (row0 + r + 8*hi), column col0+lo
    float* drow = Hout + (size_t)(row0 + 8 * hi) * H_DIM + col0 + lo;
#pragma unroll
    for (int r = 0; r < 8; ++r) drow[(size_t)r * H_DIM] = c[r];
}

// ---------------------------------------------------------------------------
// Per-node attention scores: s_src = h . a_s, s_dst = h . a_d (wave per node)
// ---------------------------------------------------------------------------
__global__ void k_scores(const float* __restrict__ Hm, const float* __restrict__ a_s,
                         const float* __restrict__ a_d, float* __restrict__ s_src,
                         float* __restrict__ s_dst) {
    int lane = threadIdx.x & 31;
    int wave = threadIdx.x >> 5;
    int node = blockIdx.x * 8 + wave;
    const float* hr = Hm + (size_t)node * H_DIM + lane * 4;
    float h0 = hr[0], h1 = hr[1], h2 = hr[2], h3 = hr[3];
    const float* as = a_s + lane * 4;
    const float* ad = a_d + lane * 4;
    float ps = h0 * as[0] + h1 * as[1] + h2 * as[2] + h3 * as[3];
    float pd = h0 * ad[0] + h1 * ad[1] + h2 * ad[2] + h3 * ad[3];
    for (int off = 16; off; off >>= 1) {
        ps += __shfl_xor(ps, off, 32);
        pd += __shfl_xor(pd, off, 32);
    }
    if (lane == 0) { s_src[node] = ps; s_dst[node] = pd; }
}

// ---------------------------------------------------------------------------
// Segment softmax + weighted aggregation, one wave per destination node.
// Self-loop handled analytically. Phase C: 512B coalesced row gathers.
// ---------------------------------------------------------------------------
__global__ void k_aggregate(const float* __restrict__ Hm,
                            const float* __restrict__ s_src,
                            const float* __restrict__ s_dst,
                            const int* __restrict__ offs,
                            const int* __restrict__ cnt,
                            const int* __restrict__ csr,
                            const float* __restrict__ bias,
                            int do_relu,
                            float* __restrict__ Xout) {
    int lane = threadIdx.x & 31;
    int wave = threadIdx.x >> 5;
    int node = blockIdx.x * 8 + wave;

    float sd = s_dst[node];
    float e_self = lrelu(s_src[node] + sd);
    int st  = offs[node];
    int deg = cnt[node];

    // Phase A: segment max (self-loop included via init)
    float m = e_self;
    for (int k = lane; k < deg; k += 32)
        m = fmaxf(m, lrelu(s_src[csr[st + k]] + sd));
    for (int off = 16; off; off >>= 1) m = fmaxf(m, __shfl_xor(m, off, 32));

    // Phase B: softmax denominator
    float dsum = 0.0f;
    for (int k = lane; k < deg; k += 32)
        dsum += __expf(lrelu(s_src[csr[st + k]] + sd) - m);
    for (int off = 16; off; off >>= 1) dsum += __shfl_xor(dsum, off, 32);
    dsum += __expf(e_self - m);
    float inv = 1.0f / (dsum + 1e-16f);

    // Phase C: weighted gather of h[src] rows (lane covers 4 channels)
    const float* hr = Hm + (size_t)node * H_DIM + lane * 4;
    float aself = __expf(e_self - m) * inv;
    float a0 = aself * hr[0], a1 = aself * hr[1], a2 = aself * hr[2], a3 = aself * hr[3];
    for (int j = 0; j < deg; ++j) {
        int s = csr[st + j];
        if (j + 1 < deg)  // pull next source row toward the WGP early
            __builtin_prefetch(Hm + (size_t)csr[st + j + 1] * H_DIM + lane * 4, 0, 0);
        float alpha = __expf(lrelu(s_src[s] + sd) - m) * inv;
        const float* hs = Hm + (size_t)s * H_DIM + lane * 4;
        a0 += alpha * hs[0];
        a1 += alpha * hs[1];
        a2 += alpha * hs[2];
        a3 += alpha * hs[3];
    }

    const float* bv = bias + lane * 4;
    a0 += bv[0]; a1 += bv[1]; a2 += bv[2]; a3 += bv[3];
    if (do_relu) {
        a0 = fmaxf(a0, 0.0f); a1 = fmaxf(a1, 0.0f);
        a2 = fmaxf(a2, 0.0f); a3 = fmaxf(a3, 0.0f);
    }
    float* xo = Xout + (size_t)node * H_DIM + lane * 4;
    xo[0] = a0; xo[1] = a1; xo[2] = a2; xo[3] = a3;
}

// ---------------------------------------------------------------------------
// Predictor: logits[64,480] = x[node0] @ pred_W + pred_b
// ---------------------------------------------------------------------------
__global__ void k_pred(const float* __restrict__ Xb, const float* __restrict__ Wp,
                       const float* __restrict__ bp, float* __restrict__ out) {
    int o = blockIdx.x * 256 + threadIdx.x;     // 30720 = 120 * 256
    int g = o / OUT_PER_GRAPH;
    int c = o - g * OUT_PER_GRAPH;
    const float* xr = Xb + (size_t)(g * NODES_PER_GRAPH) * H_DIM;
    float acc = bp[c];
#pragma unroll 4
    for (int k = 0; k < H_DIM; ++k)
        acc = fmaf(xr[k], Wp[(size_t)k * OUT_PER_GRAPH + c], acc);
    out[o] = acc;
}

// ---------------------------------------------------------------------------

extern "C" void kernel_launch(void* const* d_in, const int* in_sizes, int n_in,
                              void* d_out, int out_size, void* d_ws, size_t ws_size,
                              hipStream_t stream) {
    (void)in_sizes; (void)n_in; (void)out_size; (void)ws_size;

    const float*     x_in   = (const float*)d_in[0];
    const long long* ei     = (const long long*)d_in[1];
    /* d_in[2] = batch (sorted, NODES_PER_GRAPH each) — layout is known, unused */
    const float*     W0     = (const float*)d_in[3];
    const float*     Wsall  = (const float*)d_in[4];
    const float*     a_src  = (const float*)d_in[5];
    const float*     a_dst  = (const float*)d_in[6];
    const float*     biases = (const float*)d_in[7];
    const float*     pred_W = (const float*)d_in[8];
    const float*     pred_b = (const float*)d_in[9];
    float*           out    = (float*)d_out;

    // workspace carving (256B aligned)
    char* ws = (char*)d_ws;
    size_t off = 0;
    auto take = [&](size_t bytes) -> char* {
        char* p = ws + off;
        off = (off + bytes + 255) & ~(size_t)255;
        return p;
    };
    int*   cnt    = (int*)take((size_t)N_NODES * 4);
    int*   offs   = (int*)take((size_t)N_NODES * 4);
    int*   cursor = (int*)take((size_t)N_NODES * 4);
    int*   bsum   = (int*)take(256 * 4);
    int*   csr    = (int*)take((size_t)E_EDGES * 4);
    float* s_src  = (float*)take((size_t)N_NODES * 4);
    float* s_dst  = (float*)take((size_t)N_NODES * 4);
    float* hbuf   = (float*)take((size_t)N_NODES * H_DIM * 4);
    float* xbuf   = (float*)take((size_t)N_NODES * H_DIM * 4);

    const int nbScan  = N_NODES / 256;                 // 195 (exact)
    const int nbEdges = (E_EDGES + 255) / 256;
    const int nbNodes = (N_NODES + 255) / 256;
    const int nbGemm  = N_NODES / 16;                  // 3120
    const int nbWaveN = N_NODES / 8;                   // 6240 (8 waves/block)

    // ---- CSR build (graph is constant across hops) ----
    hipMemsetAsync(cnt,    0, (size_t)N_NODES * 4, stream);
    hipMemsetAsync(cursor, 0, (size_t)N_NODES * 4, stream);
    k_count     <<<nbEdges, 256, 0, stream>>>(ei, cnt);
    k_scan_block<<<nbScan,  256, 0, stream>>>(cnt, offs, bsum);
    k_scan_bsum <<<1,       256, 0, stream>>>(bsum, nbScan);
    k_scan_add  <<<nbScan,  256, 0, stream>>>(offs, bsum);
    k_fill      <<<nbEdges, 256, 0, stream>>>(ei, offs, cursor, csr);
    k_sortlists <<<nbNodes, 256, 0, stream>>>(offs, cnt, csr);

    // ---- 5 GAT hops ----
    for (int hop = 0; hop < HOPS; ++hop) {
        const float* X = (hop == 0) ? x_in : xbuf;
        const float* W = (hop == 0) ? W0 : (Wsall + (size_t)(hop - 1) * H_DIM * H_DIM);
        int K = (hop == 0) ? F_INDIM : H_DIM;

        k_gemm_wmma<<<nbGemm, 256, 0, stream>>>(X, W, hbuf, K);
        k_scores   <<<nbWaveN, 256, 0, stream>>>(hbuf,
                                                 a_src + (size_t)hop * H_DIM,
                                                 a_dst + (size_t)hop * H_DIM,
                                                 s_src, s_dst);
        k_aggregate<<<nbWaveN, 256, 0, stream>>>(hbuf, s_src, s_dst, offs, cnt, csr,
                                                 biases + (size_t)hop * H_DIM,
                                                 (hop < HOPS - 1) ? 1 : 0, xbuf);
    }

    // ---- predictor head ----
    k_pred<<<(B_GRAPHS * OUT_PER_GRAPH) / 256, 256, 0, stream>>>(xbuf, pred_W, pred_b, out);
}